// Advection_2791728742924
// MI455X (gfx1250) — compile-verified
//
#include <hip/hip_runtime.h>
#include <hip/hip_bf16.h>
#include <math.h>

#define GB 8
#define NP 100000
#define GS 32
#define S3 32768          // 32^3
#define STOT (GB * S3)    // 262144

typedef float v2f __attribute__((ext_vector_type(2)));
typedef float v8f __attribute__((ext_vector_type(8)));

__device__ __forceinline__ v8f wmma4(v2f a, v2f b, v8f c) {
  // D = A(16x4) * B(4x16) + C(16x16), full fp32
  return __builtin_amdgcn_wmma_f32_16x16x4_f32(false, a, false, b, (short)0, c, false, false);
}

// ---------------------------------------------------------------- zero scratch
__global__ void adv_zero(float4* __restrict__ p, int n4) {
  int i = blockIdx.x * blockDim.x + threadIdx.x;
  float4 z = {0.f, 0.f, 0.f, 0.f};
  for (; i < n4; i += gridDim.x * blockDim.x) p[i] = z;
}

// ------------------------------------------- K1: fc (1x1 conv) + P2G scatter
__global__ void adv_fc_scatter(const float* __restrict__ pc, const float* __restrict__ value,
                               const float* __restrict__ vel,
                               const float* __restrict__ fw, const float* __restrict__ fb,
                               const float* __restrict__ fg, const float* __restrict__ ft,
                               float* __restrict__ field,   // [b][s][32]
                               float* __restrict__ velg,    // [b][s][3]
                               float* __restrict__ outc) {  // concat region [b][80][N]
  int p = blockIdx.x * blockDim.x + threadIdx.x;
  if (p >= GB * NP) return;
  int b = p / NP, n = p - b * NP;

  float v[16];
#pragma unroll
  for (int c = 0; c < 16; ++c) {
    v[c] = value[(size_t)(b * 16 + c) * NP + n];
    outc[(size_t)(b * 80 + c) * NP + n] = v[c];   // concat: value passthrough
  }
  float v1[32];
#pragma unroll
  for (int o = 0; o < 32; ++o) {
    float d = fb[o];
#pragma unroll
    for (int c = 0; c < 16; ++c) d = fmaf(fw[o * 16 + c], v[c], d);
    d = fmaf(fg[o], d, ft[o]);
    d = d > 0.f ? d : 0.f;
    v1[o] = d;
    outc[(size_t)(b * 80 + 48 + o) * NP + n] = d; // concat: value1
  }

  int i0[3], i1[3]; float w1[3], w0[3];
#pragma unroll
  for (int c = 0; c < 3; ++c) {
    float f = (pc[(size_t)(b * 3 + c) * NP + n] + 1.f) * 0.5f * (float)(GS - 1);
    int a = (int)floorf(f);
    a = a < 0 ? 0 : (a > GS - 1 ? GS - 1 : a);
    i0[c] = a;
    i1[c] = (a + 1 > GS - 1) ? GS - 1 : a + 1;
    w1[c] = f - (float)a;
    w0[c] = 1.f - w1[c];
  }
  float vl[3];
#pragma unroll
  for (int c = 0; c < 3; ++c) vl[c] = vel[(size_t)(b * 3 + c) * NP + n];

#pragma unroll
  for (int dx = 0; dx < 2; ++dx)
#pragma unroll
    for (int dy = 0; dy < 2; ++dy)
#pragma unroll
      for (int dz = 0; dz < 2; ++dz) {
        int xi = dx ? i1[0] : i0[0];
        int yi = dy ? i1[1] : i0[1];
        int zi = dz ? i1[2] : i0[2];
        float w = (dx ? w1[0] : w0[0]) * (dy ? w1[1] : w0[1]) * (dz ? w1[2] : w0[2]);
        size_t base = (size_t)b * S3 + (size_t)((xi * GS + yi) * GS + zi);
        float* fgp = field + base * 32;
#pragma unroll
        for (int c = 0; c < 32; ++c) atomicAdd(fgp + c, v1[c] * w);
        float* vgp = velg + base * 3;
#pragma unroll
        for (int c = 0; c < 3; ++c) atomicAdd(vgp + c, vl[c] * w);
      }
}

// ---------------------- K2: c1 pointwise GEMM (field [b][s][c] -> h [b][c][s])
__global__ void adv_c1_gemm(const float* __restrict__ field, const float* __restrict__ W,
                            const float* __restrict__ bias, const float* __restrict__ g,
                            const float* __restrict__ bt, float* __restrict__ h) {
  const int lane = threadIdx.x & 31;
  const int lo = lane & 15, hi = lane >> 4, kofs = hi * 2;
  const int wid = blockIdx.x * (blockDim.x >> 5) + (threadIdx.x >> 5);
  const int nw = gridDim.x * (blockDim.x >> 5);

  v2f af[2][8];
  float sc[2][8], sh[2][8];
#pragma unroll
  for (int mt = 0; mt < 2; ++mt) {
#pragma unroll
    for (int ks = 0; ks < 8; ++ks)
      af[mt][ks] = *(const v2f*)(W + (mt * 16 + lo) * 32 + ks * 4 + kofs);
#pragma unroll
    for (int r = 0; r < 8; ++r) {
      int m = mt * 16 + r + hi * 8;
      sc[mt][r] = g[m];
      sh[mt][r] = fmaf(g[m], bias[m], bt[m]);
    }
  }

  const int NT = STOT / 16;
  for (int t = wid; t < NT; t += nw) {
    int s0 = t * 16;
    int b = s0 >> 15, sl = s0 & (S3 - 1);
    // prefetch next tile's B operand (gfx1250 global_prefetch)
    int tn = t + nw;
    if (tn < NT) __builtin_prefetch(field + (size_t)(tn * 16 + lo) * 32, 0, 1);
    v8f a0 = {}, a1 = {};
    const float* bp = field + (size_t)(s0 + lo) * 32 + kofs;
#pragma unroll
    for (int ks = 0; ks < 8; ++ks) {
      v2f bf = *(const v2f*)(bp + ks * 4);
      a0 = wmma4(af[0][ks], bf, a0);
      a1 = wmma4(af[1][ks], bf, a1);
    }
    float* o0 = h + ((size_t)(b * 32) << 15) + sl + lo;
#pragma unroll
    for (int r = 0; r < 8; ++r) {
      int m0 = r + hi * 8, m1 = 16 + r + hi * 8;
      float x0 = fmaf(sc[0][r], a0[r], sh[0][r]); x0 = x0 > 0.f ? x0 : 0.f;
      float x1 = fmaf(sc[1][r], a1[r], sh[1][r]); x1 = x1 > 0.f ? x1 : 0.f;
      o0[(size_t)m0 << 15] = x0;
      o0[(size_t)m1 << 15] = x1;
    }
  }
}

// ------------------- K3: c2 3x3x3 conv via 27 shifted WMMA GEMMs (h -> co)
// N-tile = 32 z-positions (full grid row) so each LDS A-fragment feeds 2 WMMAs.
__global__ void adv_c2_conv(const float* __restrict__ h, const float* __restrict__ W,
                            const float* __restrict__ bias, const float* __restrict__ g,
                            const float* __restrict__ bt, float* __restrict__ co) {
  extern __shared__ float lw[];  // [27][32*32], reordered as [off][o][k]
  for (int t = threadIdx.x; t < 32 * 32 * 27; t += blockDim.x) {
    int off = t % 27, oc = t / 27;     // oc = o*32 + c
    lw[off * 1024 + oc] = W[t];
  }
  __syncthreads();

  const int lane = threadIdx.x & 31;
  const int lo = lane & 15, hi = lane >> 4, kofs = hi * 2;
  const int wid = blockIdx.x * (blockDim.x >> 5) + (threadIdx.x >> 5);
  const int nw = gridDim.x * (blockDim.x >> 5);

  float sc[2][8], sh[2][8];
#pragma unroll
  for (int mt = 0; mt < 2; ++mt)
#pragma unroll
    for (int r = 0; r < 8; ++r) {
      int m = mt * 16 + r + hi * 8;
      sc[mt][r] = g[m];
      sh[mt][r] = fmaf(g[m], bias[m], bt[m]);
    }

  const int NT = STOT / 32;                 // one full z-row per tile
  for (int t = wid; t < NT; t += nw) {
    int s0 = t * 32;
    int b = s0 >> 15, sl = s0 & (S3 - 1);
    int x = sl >> 10, y = (sl >> 5) & 31;   // z0 == 0 for a full row
    v8f a00 = {}, a01 = {}, a10 = {}, a11 = {};
    for (int dx = -1; dx <= 1; ++dx) {
      int xx = x + dx;
      if ((unsigned)xx >= (unsigned)GS) continue;
      for (int dy = -1; dy <= 1; ++dy) {
        int yy = y + dy;
        if ((unsigned)yy >= (unsigned)GS) continue;
        const float* hb = h + (((size_t)(b * 32) + kofs) << 15) + (xx << 10) + (yy << 5);
        for (int dz = -1; dz <= 1; ++dz) {
          int z0l = lo + dz;                // N-subtile 0: z = 0..15
          int z1l = 16 + lo + dz;           // N-subtile 1: z = 16..31
          bool ok0 = (unsigned)z0l < (unsigned)GS;
          bool ok1 = (unsigned)z1l < (unsigned)GS;
          int zc0 = ok0 ? z0l : 0;
          int zc1 = ok1 ? z1l : 0;
          int offi = ((dx + 1) * 3 + (dy + 1)) * 3 + (dz + 1);
          const float* lwp = lw + offi * 1024 + kofs;
#pragma unroll
          for (int ks = 0; ks < 8; ++ks) {
            float b0x = hb[((size_t)ks << 17) + zc0];
            float b0y = hb[((size_t)ks << 17) + (1 << 15) + zc0];
            float b1x = hb[((size_t)ks << 17) + zc1];
            float b1y = hb[((size_t)ks << 17) + (1 << 15) + zc1];
            if (!ok0) { b0x = 0.f; b0y = 0.f; }
            if (!ok1) { b1x = 0.f; b1y = 0.f; }
            v2f bf0 = {b0x, b0y};
            v2f bf1 = {b1x, b1y};
            v2f w0 = *(const v2f*)(lwp + lo * 32 + ks * 4);
            v2f w1 = *(const v2f*)(lwp + (16 + lo) * 32 + ks * 4);
            a00 = wmma4(w0, bf0, a00);
            a01 = wmma4(w0, bf1, a01);
            a10 = wmma4(w1, bf0, a10);
            a11 = wmma4(w1, bf1, a11);
          }
        }
      }
    }
    float* o0 = co + ((size_t)(b * 32) << 15) + sl + lo;
#pragma unroll
    for (int r = 0; r < 8; ++r) {
      int m0 = r + hi * 8, m1 = 16 + r + hi * 8;
      float x00 = fmaf(sc[0][r], a00[r], sh[0][r]); x00 = x00 > 0.f ? x00 : 0.f;
      float x01 = fmaf(sc[0][r], a01[r], sh[0][r]); x01 = x01 > 0.f ? x01 : 0.f;
      float x10 = fmaf(sc[1][r], a10[r], sh[1][r]); x10 = x10 > 0.f ? x10 : 0.f;
      float x11 = fmaf(sc[1][r], a11[r], sh[1][r]); x11 = x11 > 0.f ? x11 : 0.f;
      o0[(size_t)m0 << 15] = x00;
      o0[((size_t)m0 << 15) + 16] = x01;
      o0[(size_t)m1 << 15] = x10;
      o0[((size_t)m1 << 15) + 16] = x11;
    }
  }
}

// ------- K4: gf + gv pointwise GEMMs (co [b][c][s] -> featg [b][s][32], nvg [b][s][3])
__global__ void adv_gfgv_gemm(const float* __restrict__ co,
                              const float* __restrict__ gfw, const float* __restrict__ gfb,
                              const float* __restrict__ gfg, const float* __restrict__ gft,
                              const float* __restrict__ gvw, const float* __restrict__ gvb,
                              float* __restrict__ featg, float* __restrict__ nvg) {
  const int lane = threadIdx.x & 31;
  const int lo = lane & 15, hi = lane >> 4, kofs = hi * 2;
  const int wid = blockIdx.x * (blockDim.x >> 5) + (threadIdx.x >> 5);
  const int nw = gridDim.x * (blockDim.x >> 5);

  v2f af[2][8], av[8];
  float sc[2][8], sh[2][8];
#pragma unroll
  for (int mt = 0; mt < 2; ++mt) {
#pragma unroll
    for (int ks = 0; ks < 8; ++ks)
      af[mt][ks] = *(const v2f*)(gfw + (mt * 16 + lo) * 32 + ks * 4 + kofs);
#pragma unroll
    for (int r = 0; r < 8; ++r) {
      int m = mt * 16 + r + hi * 8;
      sc[mt][r] = gfg[m];
      sh[mt][r] = fmaf(gfg[m], gfb[m], gft[m]);
    }
  }
#pragma unroll
  for (int ks = 0; ks < 8; ++ks) {
    int k = ks * 4 + kofs;
    float ax = (lo < 3) ? gvw[lo * 32 + k] : 0.f;
    float ay = (lo < 3) ? gvw[lo * 32 + k + 1] : 0.f;
    av[ks] = v2f{ax, ay};
  }

  const int NT = STOT / 16;
  for (int t = wid; t < NT; t += nw) {
    int s0 = t * 16;
    int b = s0 >> 15, sl = s0 & (S3 - 1);
    int tn = t + nw;
    if (tn < NT) {
      int bn = (tn * 16) >> 15, sln = (tn * 16) & (S3 - 1);
      __builtin_prefetch(co + (((size_t)(bn * 32) + kofs) << 15) + sln + lo, 0, 1);
    }
    v8f a0 = {}, a1 = {}, avc = {};
    const float* bp = co + (((size_t)(b * 32) + kofs) << 15) + sl + lo;
#pragma unroll
    for (int ks = 0; ks < 8; ++ks) {
      float bx = bp[(size_t)ks << 17];
      float by = bp[((size_t)ks << 17) + (1 << 15)];
      v2f bf = {bx, by};
      a0 = wmma4(af[0][ks], bf, a0);
      a1 = wmma4(af[1][ks], bf, a1);
      avc = wmma4(av[ks], bf, avc);
    }
    float* fo = featg + ((size_t)(s0 + lo) << 5);
#pragma unroll
    for (int r = 0; r < 8; ++r) {
      int m0 = r + hi * 8, m1 = 16 + r + hi * 8;
      float x0 = fmaf(sc[0][r], a0[r], sh[0][r]); x0 = x0 > 0.f ? x0 : 0.f;
      float x1 = fmaf(sc[1][r], a1[r], sh[1][r]); x1 = x1 > 0.f ? x1 : 0.f;
      fo[m0] = x0;
      fo[m1] = x1;
    }
    if (hi == 0) {
      float* no = nvg + (size_t)(s0 + lo) * 3;
#pragma unroll
      for (int r = 0; r < 3; ++r) no[r] = avc[r] + gvb[r];
    }
  }
}

// ---------------- K5: G2P gather + PIC/FLIP blend + outputs
__global__ void adv_gather(const float* __restrict__ pc, const float* __restrict__ vel,
                           const float* __restrict__ featg, const float* __restrict__ nvg,
                           const float* __restrict__ velg, float* __restrict__ out) {
  int p = blockIdx.x * blockDim.x + threadIdx.x;
  if (p >= GB * NP) return;
  int b = p / NP, n = p - b * NP;

  float pcv[3], vl[3];
  int i0[3], i1[3]; float w1[3], w0[3];
#pragma unroll
  for (int c = 0; c < 3; ++c) {
    pcv[c] = pc[(size_t)(b * 3 + c) * NP + n];
    vl[c] = vel[(size_t)(b * 3 + c) * NP + n];
    float f = (pcv[c] + 1.f) * 0.5f * (float)(GS - 1);
    int a = (int)floorf(f);
    a = a < 0 ? 0 : (a > GS - 1 ? GS - 1 : a);
    i0[c] = a;
    i1[c] = (a + 1 > GS - 1) ? GS - 1 : a + 1;
    w1[c] = f - (float)a;
    w0[c] = 1.f - w1[c];
  }

  float ft[32];
#pragma unroll
  for (int o = 0; o < 32; ++o) ft[o] = 0.f;
  float nv[3] = {0.f, 0.f, 0.f}, sv[3] = {0.f, 0.f, 0.f};

#pragma unroll
  for (int dx = 0; dx < 2; ++dx)
#pragma unroll
    for (int dy = 0; dy < 2; ++dy)
#pragma unroll
      for (int dz = 0; dz < 2; ++dz) {
        int xi = dx ? i1[0] : i0[0];
        int yi = dy ? i1[1] : i0[1];
        int zi = dz ? i1[2] : i0[2];
        float w = (dx ? w1[0] : w0[0]) * (dy ? w1[1] : w0[1]) * (dz ? w1[2] : w0[2]);
        size_t sn = (size_t)b * S3 + (size_t)((xi * GS + yi) * GS + zi);
        const float4* fp = (const float4*)(featg + sn * 32);
#pragma unroll
        for (int q = 0; q < 8; ++q) {
          float4 f4 = fp[q];
          ft[q * 4 + 0] = fmaf(w, f4.x, ft[q * 4 + 0]);
          ft[q * 4 + 1] = fmaf(w, f4.y, ft[q * 4 + 1]);
          ft[q * 4 + 2] = fmaf(w, f4.z, ft[q * 4 + 2]);
          ft[q * 4 + 3] = fmaf(w, f4.w, ft[q * 4 + 3]);
        }
        const float* np = nvg + sn * 3;
#pragma unroll
        for (int c = 0; c < 3; ++c) nv[c] = fmaf(w, np[c], nv[c]);
        // reference permutes the field but NOT the scattered velocity grid:
        // sampling the un-permuted scatter grid swaps x<->z indices.
        size_t ss = (size_t)b * S3 + (size_t)((zi * GS + yi) * GS + xi);
        const float* sp = velg + ss * 3;
#pragma unroll
        for (int c = 0; c < 3; ++c) sv[c] = fmaf(w, sp[c], sv[c]);
      }

  const float invG = 1.f / (float)GS;
  float* outpc = out;
  float* outft = out + (size_t)GB * 3 * NP;
  float* outvel = out + (size_t)GB * 3 * NP + (size_t)GB * 80 * NP;
#pragma unroll
  for (int c = 0; c < 3; ++c) {
    // (vel + flip)*0.3 + pic*0.7  ==  0.3*vel + sum(w*nv)/G - 0.3*sum(w*sv)/G
    float vn = 0.3f * vl[c] + nv[c] * invG - 0.3f * sv[c] * invG;
    outvel[(size_t)(b * 3 + c) * NP + n] = vn;
    outpc[(size_t)(b * 3 + c) * NP + n] = pcv[c] + vn;
  }
#pragma unroll
  for (int o = 0; o < 32; ++o) outft[(size_t)(b * 80 + 16 + o) * NP + n] = ft[o];
}

// ----------------------------------------------------------------------------
extern "C" void kernel_launch(void* const* d_in, const int* in_sizes, int n_in,
                              void* d_out, int out_size, void* d_ws, size_t ws_size,
                              hipStream_t stream) {
  (void)in_sizes; (void)n_in; (void)out_size; (void)ws_size;
  const float* pc   = (const float*)d_in[0];
  const float* value= (const float*)d_in[1];
  const float* vel  = (const float*)d_in[2];
  const float* fcw  = (const float*)d_in[3];
  const float* fcb  = (const float*)d_in[4];
  const float* fcg  = (const float*)d_in[5];
  const float* fct  = (const float*)d_in[6];
  const float* c1w  = (const float*)d_in[7];
  const float* c1b  = (const float*)d_in[8];
  const float* c1g  = (const float*)d_in[9];
  const float* c1t  = (const float*)d_in[10];
  const float* c2w  = (const float*)d_in[11];
  const float* c2b  = (const float*)d_in[12];
  const float* c2g  = (const float*)d_in[13];
  const float* c2t  = (const float*)d_in[14];
  const float* gfw  = (const float*)d_in[15];
  const float* gfb  = (const float*)d_in[16];
  const float* gfg  = (const float*)d_in[17];
  const float* gft  = (const float*)d_in[18];
  const float* gvw  = (const float*)d_in[19];
  const float* gvb  = (const float*)d_in[20];
  float* out = (float*)d_out;
  float* ws  = (float*)d_ws;

  float* field = ws;                  // 8,388,608 floats  [b][s][32]; reused as co [b][c][s]
  float* velg  = ws + 8388608;        //   786,432 floats  [b][s][3]
  float* h     = ws + 9175040;        // 8,388,608 floats  [b][c][s]; reused as featg [b][s][32]
  float* nvg   = ws + 17563648;       //   786,432 floats  [b][s][3]
  float* co    = field;
  float* featg = h;
  float* outc  = out + (size_t)GB * 3 * NP;

  // zero field + velg (contiguous 9,175,040 floats)
  adv_zero<<<4096, 256, 0, stream>>>((float4*)ws, 9175040 / 4);
  adv_fc_scatter<<<(GB * NP + 255) / 256, 256, 0, stream>>>(
      pc, value, vel, fcw, fcb, fcg, fct, field, velg, outc);
  adv_c1_gemm<<<512, 256, 0, stream>>>(field, c1w, c1b, c1g, c1t, h);
  adv_c2_conv<<<256, 256, 27648 * 4, stream>>>(h, c2w, c2b, c2g, c2t, co);
  adv_gfgv_gemm<<<512, 256, 0, stream>>>(co, gfw, gfb, gfg, gft, gvw, gvb, featg, nvg);
  adv_gather<<<(GB * NP + 255) / 256, 256, 0, stream>>>(pc, vel, featg, nvg, velg, out);
}